// TaskAlignedAssigner_81466939671039
// MI455X (gfx1250) — compile-verified
//
#include <hip/hip_runtime.h>
#include <cmath>

#define B_   16
#define N_   8400
#define C_   80
#define M_   64
#define TOPK_ 13
#define EPSF 1e-9f

// d_out layout (floats, concatenated in reference return order)
#define OFF_TB   134400        // target_bboxes  (B*N*4)
#define OFF_CLS  672000        // cls_targets    (B*N*C)
#define OFF_FG   11424000      // fg_mask        (B*N)
#define OFF_TGT  11558400      // tgt_gt_idx     (B*N)

// workspace layout
#define WS_SCORES_BYTES (16ull*64ull*8400ull*4ull)   // 34,406,400 B : scores[B][M][N] f32

typedef __attribute__((ext_vector_type(2))) float v2f;
typedef __attribute__((ext_vector_type(8))) float v8f;

// ---------------------------------------------------------------- utilities
__device__ __forceinline__ float iou_box(float lx, float ly, float rx, float ry,
                                         const float4& pb) {
  float ix = fminf(rx, pb.z) - fmaxf(lx, pb.x);
  float iy = fminf(ry, pb.w) - fmaxf(ly, pb.y);
  float inter = fmaxf(ix, 0.0f) * fmaxf(iy, 0.0f);
  float areaA = (rx - lx) * (ry - ly);
  float areaB = (pb.z - pb.x) * (pb.w - pb.y);
  return inter / fmaxf(areaA + areaB - inter, EPSF);
}

// ---------------------------------------------------------------- kernel 0: zero bits
__global__ void zero_bits_kernel(unsigned long long* bits) {
  int i = blockIdx.x * blockDim.x + threadIdx.x;
  if (i < B_ * N_) bits[i] = 0ull;
}

// ---------------------------------------------------------------- kernel 1: WMMA one-hot gather
// scores[b][m][n] = sigmoid(pd_scores[b][n][clip(label[b][m],0)])
// Computed as sigmoid(PS) (16 anchors x 80) x onehot (80 x 64) via
// V_WMMA_F32_16X16X4_F32 -- bit-exact (each dot product has exactly one
// nonzero term).
__global__ __launch_bounds__(32) void gather_wmma_kernel(const float* __restrict__ ps,
                                                         const int* __restrict__ gtl,
                                                         float* __restrict__ scores) {
  const int wid  = blockIdx.x;
  const int b    = wid / (N_ / 16);
  const int nt   = wid % (N_ / 16);
  const int lane = threadIdx.x;
  const int col   = lane & 15;   // A-row anchor / B-col m / D-col m
  const int khalf = lane >> 4;   // 0 -> K{0,1}, 1 -> K{2,3}

  // labels for this lane's m-column in each of the 4 m-tiles (clipped >= 0)
  int lm[4];
#pragma unroll
  for (int mt = 0; mt < 4; ++mt) {
    int l = gtl[b * M_ + mt * 16 + col];
    lm[mt] = l < 0 ? 0 : l;
  }

  v8f acc[4];
#pragma unroll
  for (int mt = 0; mt < 4; ++mt)
#pragma unroll
    for (int v = 0; v < 8; ++v) acc[mt][v] = 0.0f;

  const int arow = nt * 16 + col;
  const float* prow = ps + ((size_t)b * N_ + arow) * C_;

  for (int ch = 0; ch < C_ / 4; ++ch) {
    const int c0 = ch * 4 + 2 * khalf;      // this lane's two K slots
    float2 x = *(const float2*)(prow + c0); // 8B aligned (c0 even)
    v2f a;
    a.x = 1.0f / (1.0f + expf(-x.x));
    a.y = 1.0f / (1.0f + expf(-x.y));
#pragma unroll
    for (int mt = 0; mt < 4; ++mt) {
      v2f bb;
      bb.x = (lm[mt] == c0)     ? 1.0f : 0.0f;
      bb.y = (lm[mt] == c0 + 1) ? 1.0f : 0.0f;
      acc[mt] = __builtin_amdgcn_wmma_f32_16x16x4_f32(
          false, a, false, bb, (short)0, acc[mt], false, false);
    }
  }

  // D layout: VGPR v, lanes 0-15 -> anchor row v; lanes 16-31 -> row v+8.
  // Each lane stores 8 contiguous floats.
#pragma unroll
  for (int mt = 0; mt < 4; ++mt) {
    const int m = mt * 16 + col;
    float* dst = scores + ((size_t)b * M_ + m) * N_ + nt * 16 + 8 * khalf;
#pragma unroll
    for (int v = 0; v < 8; ++v) dst[v] = acc[mt][v];
  }
}

// ---------------------------------------------------------------- kernel 2: per-(b,m) top-13
__global__ __launch_bounds__(256) void topk_kernel(const float* __restrict__ pdb,
                                                   const float* __restrict__ anc,
                                                   const float* __restrict__ gtb,
                                                   const float* __restrict__ mgt,
                                                   const float* __restrict__ scores,
                                                   unsigned long long* __restrict__ bits) {
  const int bm = blockIdx.x;
  const int b = bm >> 6;
  const int m = bm & 63;
  if (mgt[b * M_ + m] <= 0.5f) return;  // invalid gt -> align==0 everywhere -> no selection

  const float4 gb = ((const float4*)gtb)[b * M_ + m];
  const float lx = gb.x, ly = gb.y, rx = gb.z, ry = gb.w;
  const int t = threadIdx.x;

  float lv[TOPK_];
  int ln[TOPK_];
#pragma unroll
  for (int j = 0; j < TOPK_; ++j) { lv[j] = -1.0f; ln[j] = 1 << 30; }

  const float* srow = scores + ((size_t)b * M_ + m) * N_;
  for (int n = t; n < N_; n += 256) {
    const float ax = anc[2 * n], ay = anc[2 * n + 1];
    const float dmin = fminf(fminf(ax - lx, ay - ly), fminf(rx - ax, ry - ay));
    if (dmin > EPSF) {
      const float4 pb = ((const float4*)pdb)[(size_t)b * N_ + n];
      const float iou = iou_box(lx, ly, rx, ry, pb);
      const float i2 = iou * iou;
      const float al = srow[n] * (i2 * i2 * i2);   // score^1 * iou^6
      if (al > lv[TOPK_ - 1]) {
        int j = TOPK_ - 1;
        while (j > 0 && al > lv[j - 1]) { lv[j] = lv[j - 1]; ln[j] = ln[j - 1]; --j; }
        lv[j] = al; ln[j] = n;
      }
    }
  }

  __shared__ float cv[256 * TOPK_];
  __shared__ int   cn[256 * TOPK_];
  __shared__ float rv[256];
  __shared__ int   rn[256];
  __shared__ int   rp[256];
  __shared__ int   stop;
#pragma unroll
  for (int j = 0; j < TOPK_; ++j) { cv[t * TOPK_ + j] = lv[j]; cn[t * TOPK_ + j] = ln[j]; }
  if (t == 0) stop = 0;
  __syncthreads();

  for (int r = 0; r < TOPK_; ++r) {
    float bv = -2.0f; int bn = 1 << 30; int bp = -1;
#pragma unroll
    for (int j = 0; j < TOPK_; ++j) {
      const int p = t * TOPK_ + j;
      const float v = cv[p]; const int nn = cn[p];
      if (v > bv || (v == bv && nn < bn)) { bv = v; bn = nn; bp = p; }
    }
    rv[t] = bv; rn[t] = bn; rp[t] = bp;
    __syncthreads();
    for (int s = 128; s > 0; s >>= 1) {
      if (t < s) {
        const float v = rv[t + s]; const int nn = rn[t + s];
        if (v > rv[t] || (v == rv[t] && nn < rn[t])) { rv[t] = v; rn[t] = nn; rp[t] = rp[t + s]; }
      }
      __syncthreads();
    }
    if (t == 0) {
      if (rv[0] > EPSF) {
        atomicOr(&bits[(size_t)b * N_ + rn[0]], 1ull << m);
        cv[rp[0]] = -2.0f;  // consume winner
      } else {
        stop = 1;           // remaining values can't pass the EPS gate
      }
    }
    __syncthreads();
    if (stop) break;
  }
}

// ---------------------------------------------------------------- kernel 3: finalize per anchor
__global__ __launch_bounds__(256) void finalize_kernel(const float* __restrict__ pdb,
                                                       const float* __restrict__ anc,
                                                       const int* __restrict__ gtl,
                                                       const float* __restrict__ gtb,
                                                       const float* __restrict__ mgt,
                                                       const float* __restrict__ scores,
                                                       const unsigned long long* __restrict__ bits,
                                                       float* __restrict__ out) {
  const int b = blockIdx.y;
  const int n = blockIdx.x * blockDim.x + threadIdx.x;

  __shared__ float4 sbox[M_];
  __shared__ int    slab[M_];
  __shared__ float  smask[M_];
  if (threadIdx.x < M_) {
    const int m = threadIdx.x;
    sbox[m]  = ((const float4*)gtb)[b * M_ + m];
    slab[m]  = gtl[b * M_ + m];
    smask[m] = mgt[b * M_ + m];
  }
  __syncthreads();
  if (n >= N_) return;

  const float4 pb = ((const float4*)pdb)[(size_t)b * N_ + n];
  const float ax = anc[2 * n], ay = anc[2 * n + 1];
  const unsigned long long W = bits[(size_t)b * N_ + n];

  int fg = 0, firstSel = -1, bestM = 0;
  float bestV = -1.0f;
  for (int m = 0; m < M_; ++m) {
    const float4 gb = sbox[m];
    const float dmin = fminf(fminf(ax - gb.x, ay - gb.y), fminf(gb.z - ax, gb.w - ay));
    const bool valid = (dmin > EPSF) && (smask[m] > 0.5f);
    float iou0 = 0.0f;
    if (valid) iou0 = iou_box(gb.x, gb.y, gb.z, gb.w, pb);
    if (iou0 > bestV) { bestV = iou0; bestM = m; }   // first-max like jnp.argmax
    if (valid && ((W >> m) & 1ull)) {
      ++fg;
      if (firstSel < 0) firstSel = m;
    }
  }

  int tgt; float maskflag; bool fgb;
  if (fg > 1)       { tgt = bestM;    maskflag = 1.0f; fgb = true;  }
  else if (fg == 1) { tgt = firstSel; maskflag = 1.0f; fgb = true;  }
  else              { tgt = 0;        maskflag = 0.0f; fgb = false; }

  float per_anchor = 0.0f;
  if (maskflag > 0.0f) {
    const float4 gb = sbox[tgt];
    const float dmin = fminf(fminf(ax - gb.x, ay - gb.y), fminf(gb.z - ax, gb.w - ay));
    const bool valid = (dmin > EPSF) && (smask[tgt] > 0.5f);
    float iou0 = 0.0f;
    if (valid) iou0 = iou_box(gb.x, gb.y, gb.z, gb.w, pb);
    const float sc = scores[((size_t)b * M_ + tgt) * N_ + n];
    const float i2 = iou0 * iou0;
    const float al = valid ? sc * (i2 * i2 * i2) : 0.0f;
    per_anchor = al / (al + EPSF) * iou0;   // align/max_align * iou * mask (single nonzero m)
  }

  const size_t idx = (size_t)b * N_ + n;
  const int lab = slab[tgt];

  out[idx] = (float)lab;                                   // target_labels
  ((float4*)(out + OFF_TB))[idx] = sbox[tgt];              // target_bboxes

  float* cls = out + OFF_CLS + idx * (size_t)C_;           // cls_targets (one-hot * per_anchor)
  const int labc = (lab >= 0 && lab < C_) ? lab : -1;
#pragma unroll
  for (int k = 0; k < C_ / 4; ++k) {
    float4 z = make_float4(0.f, 0.f, 0.f, 0.f);
    if (labc >= 0 && (labc >> 2) == k) ((float*)&z)[labc & 3] = per_anchor;
    ((float4*)cls)[k] = z;
  }

  out[OFF_FG + idx]  = fgb ? 1.0f : 0.0f;                  // fg_mask
  out[OFF_TGT + idx] = (float)tgt;                         // tgt_gt_idx
}

// ---------------------------------------------------------------- launcher
extern "C" void kernel_launch(void* const* d_in, const int* in_sizes, int n_in,
                              void* d_out, int out_size, void* d_ws, size_t ws_size,
                              hipStream_t stream) {
  const float* pd_scores = (const float*)d_in[0];
  const float* pd_bboxes = (const float*)d_in[1];
  const float* anc       = (const float*)d_in[2];
  const int*   gtl       = (const int*)d_in[3];
  const float* gtb       = (const float*)d_in[4];
  const float* mgt       = (const float*)d_in[5];
  float* out = (float*)d_out;

  float* scores = (float*)d_ws;
  unsigned long long* bits = (unsigned long long*)((char*)d_ws + WS_SCORES_BYTES);

  zero_bits_kernel<<<(B_ * N_ + 255) / 256, 256, 0, stream>>>(bits);
  gather_wmma_kernel<<<B_ * (N_ / 16), 32, 0, stream>>>(pd_scores, gtl, scores);
  topk_kernel<<<B_ * M_, 256, 0, stream>>>(pd_bboxes, anc, gtb, mgt, scores, bits);
  finalize_kernel<<<dim3((N_ + 255) / 256, B_), 256, 0, stream>>>(
      pd_bboxes, anc, gtl, gtb, mgt, scores, bits, out);
}